// QuantumKerasLayer_59811714564152
// MI455X (gfx1250) — compile-verified
//
#include <hip/hip_runtime.h>

// MI455X / gfx1250 quantum circuit simulator.
// Strategy: state vector (2^12 complex f32 = 32KB) lives entirely in LDS,
// one workgroup (256 threads / 8 waves) per batch sample. Per variational
// layer the 12 Rot gates are fused into three 16x16 complex unitaries
// (Kronecker products over 4-qubit groups) and applied with chained
// V_WMMA_F32_16X16X4_F32 ops. The 12-CNOT ring collapses into one
// precomputed basis permutation sweep. RY encoding collapses into a direct
// product-state initialization. Output = sum_i |psi_i|^2 * (12-2*popc(i))/12.

typedef __attribute__((ext_vector_type(2))) float v2f;
typedef __attribute__((ext_vector_type(8))) float v8f;

namespace {
constexpr int kNQ    = 12;          // qubits
constexpr int kDepth = 6;           // layers
constexpr int kDim   = 1 << kNQ;    // 4096 amplitudes
constexpr int kBlk   = 256;         // threads per block (8 waves of 32)
}

__global__ __launch_bounds__(kBlk) void qsim_wmma_kernel(
    const float* __restrict__ inputs,   // [B, 12] f32
    const float* __restrict__ w,        // [6, 12, 3] f32
    float* __restrict__ out)            // [B] f32
{
  __shared__ float sr[kDim];                    // state, real plane (16KB)
  __shared__ float si[kDim];                    // state, imag plane (16KB)
  __shared__ float ur[3 * 256];                 // 16x16 group unitaries, real
  __shared__ float ui[3 * 256];                 // 16x16 group unitaries, imag
  __shared__ float rmre[kDepth * kNQ * 4];      // per-qubit Rot 2x2, real
  __shared__ float rmim[kDepth * kNQ * 4];      // per-qubit Rot 2x2, imag
  __shared__ unsigned short perm[kDim];         // CNOT-ring source permutation
  __shared__ float cv[kNQ];                     // cos(x/2) encoding
  __shared__ float sv[kNQ];                     // sin(x/2) encoding
  __shared__ float red[kBlk / 32];              // per-wave reduction partials

  const int tid = threadIdx.x;
  const int b   = blockIdx.x;

  // ---- encoding angles (per sample) ----
  if (tid < kNQ) {
    const float x = 0.5f * inputs[b * kNQ + tid];
    cv[tid] = cosf(x);
    sv[tid] = sinf(x);
  }

  // ---- per-qubit Rot(phi,theta,omega) 2x2 matrices (shared weights) ----
  if (tid < kDepth * kNQ) {
    const int d = tid / kNQ;
    const int q = tid - d * kNQ;
    const float phi = w[(d * kNQ + q) * 3 + 0];
    const float th  = w[(d * kNQ + q) * 3 + 1];
    const float om  = w[(d * kNQ + q) * 3 + 2];
    const float c  = cosf(0.5f * th);
    const float s  = sinf(0.5f * th);
    const float ap = 0.5f * (phi + om);
    const float am = 0.5f * (phi - om);
    const float cp = cosf(ap), spv = sinf(ap);
    const float cm = cosf(am), smv = sinf(am);
    const int o = tid * 4;
    rmre[o + 0] =  cp * c;  rmim[o + 0] = -spv * c;   // m00 = e^{-i ap} c
    rmre[o + 1] = -cm * s;  rmim[o + 1] = -smv * s;   // m01 = -e^{+i am} s
    rmre[o + 2] =  cm * s;  rmim[o + 2] = -smv * s;   // m10 = e^{-i am} s
    rmre[o + 3] =  cp * c;  rmim[o + 3] =  spv * c;   // m11 = e^{+i ap} c
  }

  // ---- composed CNOT-ring permutation: new[j] = old[perm[j]] ----
  // Gates applied in order q=0..11: CNOT(q,(q+1)%12); compose source maps
  // in reverse. bitpos(q) = 11-q.
  for (int j = tid; j < kDim; j += kBlk) {
    int src = j;
    #pragma unroll
    for (int q = kNQ - 1; q >= 0; --q) {
      const int cpos = (kNQ - 1) - q;
      const int tpos = (kNQ - 1) - ((q + 1) % kNQ);
      src ^= ((src >> cpos) & 1) << tpos;
    }
    perm[j] = (unsigned short)src;
  }
  __syncthreads();

  // ---- product-state init: amplitude_i = prod_q (bit ? sin : cos) ----
  for (int i = tid; i < kDim; i += kBlk) {
    float p = 1.0f;
    #pragma unroll
    for (int q = 0; q < kNQ; ++q)
      p *= ((i >> ((kNQ - 1) - q)) & 1) ? sv[q] : cv[q];
    sr[i] = p;
    si[i] = 0.0f;
  }
  __syncthreads();

  const int lane = tid & 31;
  const int wv   = tid >> 5;     // wave id 0..7
  const int hf   = lane >> 4;    // wmma lane half
  const int l16  = lane & 15;
  const int rU   = tid >> 4;     // (row,col) this thread builds in U
  const int kU   = tid & 15;

  for (int d = 0; d < kDepth; ++d) {
    // ---- build three 16x16 group unitaries: U = M_{4g} (x) ... (x) M_{4g+3}
    // r/k bit (3-j) <-> qubit 4g+j (index-bit lo+3-j).
    #pragma unroll
    for (int g = 0; g < 3; ++g) {
      float are = 1.0f, aim = 0.0f;
      #pragma unroll
      for (int j = 0; j < 4; ++j) {
        const int q  = 4 * g + j;
        const int rb = (rU >> (3 - j)) & 1;
        const int kb = (kU >> (3 - j)) & 1;
        const int o  = (d * kNQ + q) * 4 + rb * 2 + kb;
        const float mre = rmre[o], mim = rmim[o];
        const float nre = are * mre - aim * mim;
        const float nim = are * mim + aim * mre;
        are = nre; aim = nim;
      }
      ur[g * 256 + rU * 16 + kU] = are;
      ui[g * 256 + rU * 16 + kU] = aim;
    }
    __syncthreads();

    // ---- apply the three fused gates: D = U * S per 16x16 column tile ----
    for (int g = 0; g < 3; ++g) {
      const int lo = 8 - 4 * g;   // bit offset of the 4-bit group field
      #pragma unroll
      for (int tt = 0; tt < 2; ++tt) {
        const int tile = wv + tt * 8;          // 16 tiles, wave-disjoint
        const int o    = tile * 16 + l16;      // this lane's column (outer idx)
        const int base = ((o >> lo) << (lo + 4)) | (o & ((1 << lo) - 1));

        // Fragment loads per ISA layouts.
        // A 16x4 f32: lanes0-15 M=l16, VGPR0=K, VGPR1=K+1; lanes16-31: K+2.
        // B 4x16 f32: mirrors A with N on lanes 0-15.
        v2f br[4], bi[4], ar[4], ai[4], an[4];
        #pragma unroll
        for (int c = 0; c < 4; ++c) {
          const int k0 = 4 * c + 2 * hf;
          const int i0 = base + (k0 << lo);
          const int i1 = base + ((k0 + 1) << lo);
          br[c].x = sr[i0]; br[c].y = sr[i1];
          bi[c].x = si[i0]; bi[c].y = si[i1];
          ar[c].x = ur[g * 256 + l16 * 16 + k0];
          ar[c].y = ur[g * 256 + l16 * 16 + k0 + 1];
          ai[c].x = ui[g * 256 + l16 * 16 + k0];
          ai[c].y = ui[g * 256 + l16 * 16 + k0 + 1];
          an[c].x = -ai[c].x;  an[c].y = -ai[c].y;   // for -Ui*Si term
        }

        v8f dr = {};
        v8f di = {};
#if __has_builtin(__builtin_amdgcn_wmma_f32_16x16x4_f32)
        // Dr = Ur*Sr - Ui*Si ; Di = Ur*Si + Ui*Sr   (K=16 via 4 chained WMMAs)
        #pragma unroll
        for (int c = 0; c < 4; ++c)
          dr = __builtin_amdgcn_wmma_f32_16x16x4_f32(false, ar[c], false, br[c],
                                                     (short)0, dr, false, false);
        #pragma unroll
        for (int c = 0; c < 4; ++c)
          dr = __builtin_amdgcn_wmma_f32_16x16x4_f32(false, an[c], false, bi[c],
                                                     (short)0, dr, false, false);
        #pragma unroll
        for (int c = 0; c < 4; ++c)
          di = __builtin_amdgcn_wmma_f32_16x16x4_f32(false, ar[c], false, bi[c],
                                                     (short)0, di, false, false);
        #pragma unroll
        for (int c = 0; c < 4; ++c)
          di = __builtin_amdgcn_wmma_f32_16x16x4_f32(false, ai[c], false, br[c],
                                                     (short)0, di, false, false);
#else
        // VALU fallback (host pass / builtin-missing): read all S first
        // (lockstep within wave keeps reads ahead of the writes below).
        float s0v[16], s1v[16];
        #pragma unroll
        for (int k = 0; k < 16; ++k) {
          s0v[k] = sr[base + (k << lo)];
          s1v[k] = si[base + (k << lo)];
        }
        #pragma unroll
        for (int rr = 0; rr < 8; ++rr) {
          const int m = rr + 8 * hf;
          float accr = 0.0f, acci = 0.0f;
          #pragma unroll
          for (int k = 0; k < 16; ++k) {
            const float ure = ur[g * 256 + m * 16 + k];
            const float uim = ui[g * 256 + m * 16 + k];
            accr += ure * s0v[k] - uim * s1v[k];
            acci += ure * s1v[k] + uim * s0v[k];
          }
          dr[rr] = accr; di[rr] = acci;
        }
#endif
        // D scatter: lane holds N=l16, rows M = rr + 8*hf.
        #pragma unroll
        for (int rr = 0; rr < 8; ++rr) {
          const int m = rr + 8 * hf;
          sr[base + (m << lo)] = dr[rr];
          si[base + (m << lo)] = di[rr];
        }
      }
      __syncthreads();
    }

    // ---- CNOT ring as one permutation sweep (read all, barrier, write) ----
    {
      float tr[16], tiv[16];
      #pragma unroll
      for (int ii = 0; ii < 16; ++ii) {
        const int j = ii * kBlk + tid;
        const int s = perm[j];
        tr[ii] = sr[s]; tiv[ii] = si[s];
      }
      __syncthreads();
      #pragma unroll
      for (int ii = 0; ii < 16; ++ii) {
        const int j = ii * kBlk + tid;
        sr[j] = tr[ii]; si[j] = tiv[ii];
      }
      __syncthreads();
    }
  }

  // ---- mean_q <Z_q> = sum_i p_i * (12 - 2*popc(i)) / 12 ----
  float acc = 0.0f;
  for (int i = tid; i < kDim; i += kBlk) {
    const float a = sr[i];
    const float c = si[i];
    acc += (a * a + c * c) * (float)(kNQ - 2 * __popc(i));
  }
  acc *= (1.0f / (float)kNQ);
  #pragma unroll
  for (int off = 16; off > 0; off >>= 1)
    acc += __shfl_down(acc, off, 32);
  if (lane == 0) red[wv] = acc;
  __syncthreads();
  if (tid == 0) {
    float t = 0.0f;
    #pragma unroll
    for (int i = 0; i < kBlk / 32; ++i) t += red[i];
    out[b] = t;
  }
}

extern "C" void kernel_launch(void* const* d_in, const int* in_sizes, int n_in,
                              void* d_out, int out_size, void* d_ws, size_t ws_size,
                              hipStream_t stream) {
  const float* inputs = (const float*)d_in[0];   // [B, 12] f32
  const float* w      = (const float*)d_in[1];   // [6, 12, 3] f32
  float* out          = (float*)d_out;           // [B] f32
  const int B = in_sizes[0] / kNQ;               // 4096
  qsim_wmma_kernel<<<B, kBlk, 0, stream>>>(inputs, w, out);
}